// MyModel_89833535963284
// MI455X (gfx1250) — compile-verified
//
#include <hip/hip_runtime.h>
#include <math.h>

// ---------------------------------------------------------------------------
// GCN + TopKPool pipeline for MI455X (gfx1250, wave32).
// GEMMs use V_WMMA_F32_16X16X4_F32 (exact fp32, matches reference numerics).
// Scatter aggregation via global f32 atomics (resolved in 192MB L2).
// ---------------------------------------------------------------------------

typedef float v2f __attribute__((ext_vector_type(2)));
typedef float v8f __attribute__((ext_vector_type(8)));

#define HID 64
#define EPS 1e-5f

constexpr int B_   = 128;
constexpr int NPG  = 1024;
constexpr int DEG8 = 8;
constexpr int OUTD = 10;
constexpr int N0   = B_ * NPG;      // 131072 nodes, layer-1 input
constexpr int NE   = N0 * DEG8;     // 1048576 edges (fixed, masked by ew)
constexpr int KP1  = 820;           // ceil(0.8*1024)
constexpr int KP2  = 656;           // ceil(0.8*820)
constexpr int KP3  = 525;           // ceil(0.8*656)
constexpr int N1   = B_ * KP1;      // 104960 (largest pooled node count)

// ---- workspace layout (bytes, all 256B-aligned by construction) -----------
constexpr size_t SZ_XW   = (size_t)N0 * HID * 4;   // 33554432
constexpr size_t SZ_X    = (size_t)N1 * HID * 4;   // 26869760
constexpr size_t SZ_NODE = (size_t)N0 * 4;         // 524288
constexpr size_t SZ_EDGE = (size_t)NE * 4;         // 4194304
constexpr size_t OFF_XW    = 0;
constexpr size_t OFF_AGG   = OFF_XW   + SZ_XW;
constexpr size_t OFF_X     = OFF_AGG  + SZ_XW;
constexpr size_t OFF_DEG   = OFF_X    + SZ_X;
constexpr size_t OFF_DINV  = OFF_DEG  + SZ_NODE;
constexpr size_t OFF_SCORE = OFF_DINV + SZ_NODE;
constexpr size_t OFF_MAP   = OFF_SCORE+ SZ_NODE;
constexpr size_t OFF_PERM  = OFF_MAP  + SZ_NODE;
constexpr size_t OFF_SRC   = OFF_PERM + (size_t)N1 * 4;
constexpr size_t OFF_DST   = OFF_SRC  + SZ_EDGE;
constexpr size_t OFF_EW    = OFF_DST  + SZ_EDGE;
constexpr size_t OFF_STAT  = OFF_EW   + SZ_EDGE;   // 256 floats: sum|sumsq|mu|scale
constexpr size_t OFF_RO    = OFF_STAT + 1024;      // B x 2*HID readout

// ---------------------------------------------------------------------------
// GEMM: Y[M,64] = X[M,K] @ W[K,64], one wave per 16-row block, 16x64 tile.
// A fragment: lanes 0-15 -> K={k,k+1}, lanes 16-31 -> K={k+2,k+3} (row = l16).
// C/D: VGPR r -> rows r (lanes 0-15) and r+8 (lanes 16-31), col = l16.
// No divergence anywhere (WMMA requires EXEC all-ones).
// ---------------------------------------------------------------------------
__global__ void __launch_bounds__(32)
k_gemm16x64(const float* __restrict__ X, const float* __restrict__ W,
            float* __restrict__ Y, int K) {
  const int lane = threadIdx.x;
  const int half = lane >> 4;
  const int l16  = lane & 15;
  const float* __restrict__ xr = X + ((size_t)blockIdx.x * 16 + l16) * K;
  v8f a0 = {}, a1 = {}, a2 = {}, a3 = {};
  for (int k = 0; k < K; k += 4) {
    v2f a;
    a.x = xr[k + half * 2];
    a.y = xr[k + half * 2 + 1];
    const float* bp = W + (size_t)(k + half * 2) * HID + l16;
    v2f b0, b1, b2, b3;
    b0.x = bp[0];  b0.y = bp[HID];
    b1.x = bp[16]; b1.y = bp[HID + 16];
    b2.x = bp[32]; b2.y = bp[HID + 32];
    b3.x = bp[48]; b3.y = bp[HID + 48];
    a0 = __builtin_amdgcn_wmma_f32_16x16x4_f32(false, a, false, b0, (short)0, a0, false, false);
    a1 = __builtin_amdgcn_wmma_f32_16x16x4_f32(false, a, false, b1, (short)0, a1, false, false);
    a2 = __builtin_amdgcn_wmma_f32_16x16x4_f32(false, a, false, b2, (short)0, a2, false, false);
    a3 = __builtin_amdgcn_wmma_f32_16x16x4_f32(false, a, false, b3, (short)0, a3, false, false);
  }
  float* yp = Y + ((size_t)blockIdx.x * 16 + half * 8) * HID + l16;
#pragma unroll
  for (int r = 0; r < 8; ++r) {
    yp[(size_t)r * HID + 0]  = a0[r];
    yp[(size_t)r * HID + 16] = a1[r];
    yp[(size_t)r * HID + 32] = a2[r];
    yp[(size_t)r * HID + 48] = a3[r];
  }
}

// ---- utility kernels ------------------------------------------------------
__global__ void k_zero(float* p, int n) {
  for (int i = blockIdx.x * blockDim.x + threadIdx.x; i < n; i += gridDim.x * blockDim.x)
    p[i] = 0.f;
}
__global__ void k_fill_neg1(int* p, int n) {
  int i = blockIdx.x * blockDim.x + threadIdx.x;
  if (i < n) p[i] = -1;
}
__global__ void k_edge_init(const int* __restrict__ ei, int* src, int* dst, float* ew) {
  int e = blockIdx.x * blockDim.x + threadIdx.x;
  src[e] = ei[e];
  dst[e] = ei[NE + e];
  ew[e]  = 1.f;
}

// ---- degree + symmetric normalization -------------------------------------
__global__ void k_edge_deg(const int* __restrict__ dst, const float* __restrict__ ew,
                           float* deg) {
  int e = blockIdx.x * blockDim.x + threadIdx.x;
  float w = ew[e];
  if (w != 0.f) atomicAdd(&deg[dst[e]], w);
}
__global__ void k_dinv(const float* __restrict__ deg, float* dinv, int n) {
  int i = blockIdx.x * blockDim.x + threadIdx.x;
  if (i < n) dinv[i] = rsqrtf(deg[i] + 1.0f);   // +1 self loop
}

// agg = xw * dinv^2 + b  (self-loop term + bias)
__global__ void k_self_bias(const float* __restrict__ xw, const float* __restrict__ dinv,
                            const float* __restrict__ bias, float* agg) {
  int t = blockIdx.x * blockDim.x + threadIdx.x;
  int i = t >> 6, f = t & 63;
  float di = dinv[i];
  agg[t] = xw[t] * di * di + bias[f];
}

// agg[dst] += xw[src] * dinv[src]*dinv[dst]*ew   (8 features per thread)
__global__ void k_edge_agg(const int* __restrict__ src, const int* __restrict__ dst,
                           const float* __restrict__ ew, const float* __restrict__ dinv,
                           const float* __restrict__ xw, float* agg) {
  int t = blockIdx.x * blockDim.x + threadIdx.x;
  int e = t >> 3, f0 = (t & 7) * 8;
  float w = ew[e];
  if (w != 0.f) {
    int s = src[e], d = dst[e];
    float nrm = dinv[s] * dinv[d] * w;
    const float* xs = xw + (size_t)s * HID + f0;
    float* ad = agg + (size_t)d * HID + f0;
#pragma unroll
    for (int f = 0; f < 8; ++f) atomicAdd(&ad[f], xs[f] * nrm);
  }
}

// ---- BatchNorm (training stats over node dim) + ReLU ----------------------
__global__ void k_bn_sums(const float* __restrict__ h, float* stat, int n) {
  __shared__ float ls[4][64], lq[4][64];
  int f = threadIdx.x & 63, r = threadIdx.x >> 6;
  float s = 0.f, q = 0.f;
  for (int i = blockIdx.x * 4 + r; i < n; i += gridDim.x * 4) {
    float v = h[(size_t)i * HID + f];
    s += v; q += v * v;
  }
  ls[r][f] = s; lq[r][f] = q;
  __syncthreads();
  if (r == 0) {
    s = ls[0][f] + ls[1][f] + ls[2][f] + ls[3][f];
    q = lq[0][f] + lq[1][f] + lq[2][f] + lq[3][f];
    atomicAdd(&stat[f], s);
    atomicAdd(&stat[64 + f], q);
  }
}
__global__ void k_bn_finalize(float* stat, const float* __restrict__ gamma, int n) {
  int f = threadIdx.x;
  float inv_n = 1.f / (float)n;
  float mu  = stat[f] * inv_n;
  float var = stat[64 + f] * inv_n - mu * mu;   // biased variance
  stat[128 + f] = mu;
  stat[192 + f] = gamma[f] * rsqrtf(var + EPS);
}
__global__ void k_bn_apply(float* h, const float* __restrict__ stat,
                           const float* __restrict__ beta) {
  int t = blockIdx.x * blockDim.x + threadIdx.x;
  int f = t & 63;
  float v = (h[t] - stat[128 + f]) * stat[192 + f] + beta[f];
  h[t] = fmaxf(v, 0.f);
}

// ---- TopK pooling ---------------------------------------------------------
__global__ void k_score(const float* __restrict__ h, const float* __restrict__ p,
                        float* score, int n) {
  int i = blockIdx.x * blockDim.x + threadIdx.x;
  if (i >= n) return;
  float dot = 0.f, pn = 0.f;
  const float* hr = h + (size_t)i * HID;
#pragma unroll
  for (int f = 0; f < HID; ++f) { dot += hr[f] * p[f]; pn += p[f] * p[f]; }
  score[i] = dot * rsqrtf(pn);
}

// One block per graph: rank-select (stable, tie-break by lower index).
__global__ void __launch_bounds__(1024)
k_topk(const float* __restrict__ score, int* perm, int* mapping, int n, int k) {
  __shared__ float ss[1024];
  int g = blockIdx.x, t = threadIdx.x;
  if (t < n) ss[t] = score[(size_t)g * n + t];
  __syncthreads();
  if (t < n) {
    float mys = ss[t];
    int rank = 0;
    for (int j = 0; j < n; ++j) {
      float sj = ss[j];
      rank += (sj > mys) || (sj == mys && j < t);
    }
    if (rank < k) {
      int ni = g * k + rank;
      perm[ni] = g * n + t;
      mapping[g * n + t] = ni;
    }
  }
}

__global__ void k_gather(const float* __restrict__ h, const float* __restrict__ score,
                         const int* __restrict__ perm, float* xn) {
  int t = blockIdx.x * blockDim.x + threadIdx.x;
  int i = t >> 6, f = t & 63;
  int old = perm[i];
  xn[t] = h[(size_t)old * HID + f] * tanhf(score[old]);
}

__global__ void k_reindex(int* src, int* dst, float* ew, const int* __restrict__ mapping) {
  int e = blockIdx.x * blockDim.x + threadIdx.x;
  int s = src[e], d = dst[e];
  float w = ew[e];
  int ns = mapping[s], nd = mapping[d];
  bool valid = (ns >= 0) && (nd >= 0) && (w > 0.f);
  src[e] = valid ? ns : 0;
  dst[e] = valid ? nd : 0;
  ew[e]  = valid ? 1.f : 0.f;
}

// ---- readout: concat(mean, max) per graph, accumulated over layers --------
__global__ void k_readout(const float* __restrict__ x, float* ro, int k) {
  int t = blockIdx.x * blockDim.x + threadIdx.x;   // B*64 threads
  int g = t >> 6, f = t & 63;
  float s = 0.f, m = -INFINITY;
  const float* xg = x + (size_t)g * k * HID + f;
  for (int i = 0; i < k; ++i) {
    float v = xg[(size_t)i * HID];
    s += v;
    m = fmaxf(m, v);
  }
  ro[g * 128 + f]      += s / (float)k;
  ro[g * 128 + 64 + f] += m;
}

__global__ void k_final(const float* __restrict__ ro, const float* __restrict__ Wm,
                        const float* __restrict__ bm, float* out) {
  int t = blockIdx.x * blockDim.x + threadIdx.x;
  if (t >= B_ * OUTD) return;
  int g = t / OUTD, c = t % OUTD;
  float s = bm[c];
#pragma unroll 4
  for (int f = 0; f < 2 * HID; ++f) s += ro[g * 128 + f] * Wm[f * OUTD + c];
  out[t] = s;
}

// ---------------------------------------------------------------------------
extern "C" void kernel_launch(void* const* d_in, const int* in_sizes, int n_in,
                              void* d_out, int out_size, void* d_ws, size_t ws_size,
                              hipStream_t stream) {
  (void)in_sizes; (void)n_in; (void)out_size; (void)ws_size;
  const float* x0 = (const float*)d_in[0];
  const int*   ei = (const int*)d_in[1];
  const float* W[3]  = {(const float*)d_in[2],  (const float*)d_in[7],  (const float*)d_in[12]};
  const float* bb[3] = {(const float*)d_in[3],  (const float*)d_in[8],  (const float*)d_in[13]};
  const float* gm[3] = {(const float*)d_in[4],  (const float*)d_in[9],  (const float*)d_in[14]};
  const float* bt[3] = {(const float*)d_in[5],  (const float*)d_in[10], (const float*)d_in[15]};
  const float* pp[3] = {(const float*)d_in[6],  (const float*)d_in[11], (const float*)d_in[16]};
  const float* Wm = (const float*)d_in[17];
  const float* bm = (const float*)d_in[18];

  char* ws = (char*)d_ws;
  float* xw      = (float*)(ws + OFF_XW);
  float* agg     = (float*)(ws + OFF_AGG);
  float* xbuf    = (float*)(ws + OFF_X);
  float* deg     = (float*)(ws + OFF_DEG);
  float* dinv    = (float*)(ws + OFF_DINV);
  float* score   = (float*)(ws + OFF_SCORE);
  int*   mapping = (int*)  (ws + OFF_MAP);
  int*   perm    = (int*)  (ws + OFF_PERM);
  int*   src     = (int*)  (ws + OFF_SRC);
  int*   dst     = (int*)  (ws + OFF_DST);
  float* ew      = (float*)(ws + OFF_EW);
  float* stat    = (float*)(ws + OFF_STAT);
  float* ro      = (float*)(ws + OFF_RO);

  const int npg[4]  = {NPG, KP1, KP2, KP3};
  const int Kdim[3] = {128, HID, HID};

  k_edge_init<<<NE / 256, 256, 0, stream>>>(ei, src, dst, ew);
  k_zero<<<(B_ * 128 + 255) / 256, 256, 0, stream>>>(ro, B_ * 128);

  const float* xin = x0;
  for (int L = 0; L < 3; ++L) {
    const int n = npg[L], k = npg[L + 1];
    const int Nc = B_ * n, Nk = B_ * k, Kd = Kdim[L];

    // GCNConv
    k_gemm16x64<<<Nc / 16, 32, 0, stream>>>(xin, W[L], xw, Kd);
    k_zero<<<(Nc + 255) / 256, 256, 0, stream>>>(deg, Nc);
    k_edge_deg<<<NE / 256, 256, 0, stream>>>(dst, ew, deg);
    k_dinv<<<(Nc + 255) / 256, 256, 0, stream>>>(deg, dinv, Nc);
    k_self_bias<<<(Nc * HID) / 256, 256, 0, stream>>>(xw, dinv, bb[L], agg);
    k_edge_agg<<<(NE * 8) / 256, 256, 0, stream>>>(src, dst, ew, dinv, xw, agg);

    // BatchNorm + ReLU (in place on agg)
    k_zero<<<1, 128, 0, stream>>>(stat, 128);
    k_bn_sums<<<256, 256, 0, stream>>>(agg, stat, Nc);
    k_bn_finalize<<<1, 64, 0, stream>>>(stat, gm[L], Nc);
    k_bn_apply<<<(Nc * HID) / 256, 256, 0, stream>>>(agg, stat, bt[L]);

    // TopK pooling
    k_score<<<(Nc + 255) / 256, 256, 0, stream>>>(agg, pp[L], score, Nc);
    k_fill_neg1<<<(Nc + 255) / 256, 256, 0, stream>>>(mapping, Nc);
    k_topk<<<B_, 1024, 0, stream>>>(score, perm, mapping, n, k);
    k_gather<<<(Nk * HID) / 256, 256, 0, stream>>>(agg, score, perm, xbuf);
    k_reindex<<<NE / 256, 256, 0, stream>>>(src, dst, ew, mapping);

    // readout accumulation
    k_readout<<<(B_ * HID) / 256, 256, 0, stream>>>(xbuf, ro, k);

    xin = xbuf;
  }

  k_final<<<(B_ * OUTD + 255) / 256, 256, 0, stream>>>(ro, Wm, bm, (float*)d_out);
}